// QNNLayer_47991964565825
// MI455X (gfx1250) — compile-verified
//
#include <hip/hip_runtime.h>
#include <hip/hip_bf16.h>

// Strategy:
//  out[b,i] = sum_k sign_i(k) * |f_k(b)|^2,  f(b) = V * r(b)
//  where r(b) is the REAL separable embedding vector and V = U * diag((-i)^popc(j))
//  Kernel 1 (build_V): simulate the weight circuit on 256 basis columns -> V_re/V_im in d_ws (512 KB).
//  Kernel 2 (qnn_wmma): R-tiles in LDS, fp32 WMMA 16x16x4 GEMM vs V (L2-resident),
//  sign-FMA + shfl_xor reduction to <Z_i> directly (no atomics, every output written).

#define NQ 8
#define DIM 256
#define WAVES 4
#define MT (WAVES * 16)   // 64 batch rows per workgroup

typedef float v2f __attribute__((ext_vector_type(2)));
typedef float v8f __attribute__((ext_vector_type(8)));

struct C2 { float x, y; };

// ---------------- Kernel 1: build V = (weight-layer unitary) * phase ----------------
__global__ __launch_bounds__(128) void build_V(const float* __restrict__ W,
                                               float* __restrict__ Vre,
                                               float* __restrict__ Vim) {
  __shared__ C2 st[DIM];
  const int t = threadIdx.x;
  const int j = blockIdx.x;           // basis column
  st[t]       = (t == j)       ? C2{1.f, 0.f} : C2{0.f, 0.f};
  st[t + 128] = (t + 128 == j) ? C2{1.f, 0.f} : C2{0.f, 0.f};
  __syncthreads();

  for (int l = 0; l < 2; ++l) {
    for (int q = 0; q < NQ; ++q) {
      const int p  = 7 - q;                                  // wire q lives in bit (7-q)
      const int k0 = ((t >> p) << (p + 1)) | (t & ((1 << p) - 1));
      const int k1 = k0 | (1 << p);
      {  // RX: [[c, -i s], [-i s, c]]
        float th = 0.5f * W[(l * NQ + q) * 3 + 0];
        float c = cosf(th), s = sinf(th);
        C2 s0 = st[k0], s1 = st[k1];
        st[k0] = { c * s0.x + s * s1.y,  c * s0.y - s * s1.x };
        st[k1] = { s * s0.y + c * s1.x, -s * s0.x + c * s1.y };
        __syncthreads();
      }
      {  // RY: [[c, -s], [s, c]]
        float th = 0.5f * W[(l * NQ + q) * 3 + 1];
        float c = cosf(th), s = sinf(th);
        C2 s0 = st[k0], s1 = st[k1];
        st[k0] = { c * s0.x - s * s1.x, c * s0.y - s * s1.y };
        st[k1] = { s * s0.x + c * s1.x, s * s0.y + c * s1.y };
        __syncthreads();
      }
      {  // RZ: diag(c - i s, c + i s)
        float th = 0.5f * W[(l * NQ + q) * 3 + 2];
        float c = cosf(th), s = sinf(th);
        C2 s0 = st[k0], s1 = st[k1];
        st[k0] = { c * s0.x + s * s0.y, c * s0.y - s * s0.x };
        st[k1] = { c * s1.x - s * s1.y, c * s1.y + s * s1.x };
        __syncthreads();
      }
    }
    for (int q = 0; q < NQ; ++q) {     // CNOT(q -> (q+1)%8)
      const int pc = 7 - q;
      const int pg = 7 - ((q + 1) & 7);
      if (t < 64) {
        int lo = pc < pg ? pc : pg, hi = pc < pg ? pg : pc;
        int k = ((t >> lo) << (lo + 1)) | (t & ((1 << lo) - 1));
        k = ((k >> hi) << (hi + 1)) | (k & ((1 << hi) - 1));
        int a = k | (1 << pc);         // ctrl=1, tgt=0
        int b = a | (1 << pg);         // ctrl=1, tgt=1
        C2 tmp = st[a]; st[a] = st[b]; st[b] = tmp;
      }
      __syncthreads();
    }
  }

  // multiply column by (-i)^popcount(j); store V row-major as V[n][j]
  const int pc4 = __popc(j) & 3;
  for (int n = t; n < DIM; n += 128) {
    C2 a = st[n];
    float re, im;
    switch (pc4) {
      case 0:  re =  a.x; im =  a.y; break;
      case 1:  re =  a.y; im = -a.x; break;
      case 2:  re = -a.x; im = -a.y; break;
      default: re = -a.y; im =  a.x; break;
    }
    Vre[n * DIM + j] = re;
    Vim[n * DIM + j] = im;
  }
}

// ---------------- Kernel 2: WMMA GEMM + in-wave <Z> reduction ----------------
__global__ __launch_bounds__(WAVES * 32) void qnn_wmma(const float* __restrict__ X,
                                                       const float* __restrict__ Vre,
                                                       const float* __restrict__ Vim,
                                                       float* __restrict__ out, int B) {
  __shared__ float Rt[WAVES][16][DIM];        // 64 KB: per-wave 16x256 embedding tile
  const int tid = threadIdx.x;
  const int w   = tid >> 5;                   // wave id (own m-tile)
  const int l   = tid & 31;                   // lane
  const int h   = l >> 4;                     // lane half (selects K pair in frags)
  const int n4  = l & 15;                     // A-row / B-col within tile
  const int rowBase = blockIdx.x * MT + w * 16;

  // Build separable R tile: lane handles row r = l>>1, k-half = l&1
  {
    const int r = l >> 1;
    const int khalf = l & 1;
    int xr = rowBase + r; if (xr >= B) xr = B - 1;
    float fc[8], fs[8];
    #pragma unroll
    for (int q = 0; q < 8; ++q) {
      float th = 0.5f * X[xr * 8 + q];
      fc[q] = cosf(th); fs[q] = sinf(th);
    }
    float H[8], L[16];                        // k = khalf*128 + hh*16 + m
    #pragma unroll
    for (int hh = 0; hh < 8; ++hh) {
      float v = khalf ? fs[0] : fc[0];        // wire0 <-> bit7
      v *= ((hh >> 2) & 1) ? fs[1] : fc[1];
      v *= ((hh >> 1) & 1) ? fs[2] : fc[2];
      v *= (hh & 1)        ? fs[3] : fc[3];
      H[hh] = v;
    }
    #pragma unroll
    for (int m = 0; m < 16; ++m) {
      float v = ((m >> 3) & 1) ? fs[4] : fc[4];
      v *= ((m >> 2) & 1) ? fs[5] : fc[5];
      v *= ((m >> 1) & 1) ? fs[6] : fc[6];
      v *= (m & 1)        ? fs[7] : fc[7];
      L[m] = v;
    }
    #pragma unroll
    for (int hh = 0; hh < 8; ++hh)
      #pragma unroll
      for (int m = 0; m < 16; ++m)
        Rt[w][r][khalf * 128 + hh * 16 + m] = H[hh] * L[m];
  }
  __syncthreads();

  // per-lane signs, wires 4..7 (k bits 3..0 = n4 bits 3..0)
  const float sgL0 = ((n4 >> 3) & 1) ? -1.f : 1.f;
  const float sgL1 = ((n4 >> 2) & 1) ? -1.f : 1.f;
  const float sgL2 = ((n4 >> 1) & 1) ? -1.f : 1.f;
  const float sgL3 = (n4 & 1)        ? -1.f : 1.f;

  float acc[8][8];                            // [wire][D-vgpr row]
  #pragma unroll
  for (int i = 0; i < 8; ++i)
    #pragma unroll
    for (int v = 0; v < 8; ++v) acc[i][v] = 0.f;

  const float* Arow = &Rt[w][n4][0];          // A frag: m = lane%16

  for (int t = 0; t < 16; ++t) {              // 16 n-tiles = all 256 output amplitudes
    v8f dre = {}; v8f dim_ = {};
    const float* brP = Vre + (t * 16 + n4) * DIM;  // B frag: n = lane%16
    const float* biP = Vim + (t * 16 + n4) * DIM;
    #pragma unroll 4
    for (int kc = 0; kc < DIM; kc += 4) {
      v2f a  = *(const v2f*)(Arow + kc + 2 * h);   // K = kc+2h, kc+2h+1
      v2f br = *(const v2f*)(brP  + kc + 2 * h);
      v2f bi = *(const v2f*)(biP  + kc + 2 * h);
      dre  = __builtin_amdgcn_wmma_f32_16x16x4_f32(false, a, false, br, (short)0, dre,  false, false);
      dim_ = __builtin_amdgcn_wmma_f32_16x16x4_f32(false, a, false, bi, (short)0, dim_, false, false);
    }
    const float sgT0 = ((t >> 3) & 1) ? -1.f : 1.f;   // wire0 <-> k bit7
    const float sgT1 = ((t >> 2) & 1) ? -1.f : 1.f;
    const float sgT2 = ((t >> 1) & 1) ? -1.f : 1.f;
    const float sgT3 = (t & 1)        ? -1.f : 1.f;
    #pragma unroll
    for (int v = 0; v < 8; ++v) {
      float p = dre[v] * dre[v] + dim_[v] * dim_[v];  // prob of basis state k
      acc[0][v] = fmaf(sgT0, p, acc[0][v]);
      acc[1][v] = fmaf(sgT1, p, acc[1][v]);
      acc[2][v] = fmaf(sgT2, p, acc[2][v]);
      acc[3][v] = fmaf(sgT3, p, acc[3][v]);
      acc[4][v] = fmaf(sgL0, p, acc[4][v]);
      acc[5][v] = fmaf(sgL1, p, acc[5][v]);
      acc[6][v] = fmaf(sgL2, p, acc[6][v]);
      acc[7][v] = fmaf(sgL3, p, acc[7][v]);
    }
  }

  // reduce across the 16 lanes of each half (rows live at m = v + 8h)
  #pragma unroll
  for (int i = 0; i < 8; ++i)
    #pragma unroll
    for (int v = 0; v < 8; ++v) {
      float s = acc[i][v];
      s += __shfl_xor(s, 1, 32);
      s += __shfl_xor(s, 2, 32);
      s += __shfl_xor(s, 4, 32);
      s += __shfl_xor(s, 8, 32);
      acc[i][v] = s;
    }

  if (n4 == 0) {
    #pragma unroll
    for (int v = 0; v < 8; ++v) {
      int row = rowBase + h * 8 + v;
      if (row < B) {
        #pragma unroll
        for (int i = 0; i < 8; ++i)
          out[row * 8 + i] = acc[i][v];
      }
    }
  }
}

extern "C" void kernel_launch(void* const* d_in, const int* in_sizes, int n_in,
                              void* d_out, int out_size, void* d_ws, size_t ws_size,
                              hipStream_t stream) {
  const float* x = (const float*)d_in[0];     // (B, 8)
  const float* w = (const float*)d_in[1];     // (2, 8, 3)
  float* out = (float*)d_out;                 // (B, 8)
  float* Vre = (float*)d_ws;                  // 256*256 f32
  float* Vim = Vre + DIM * DIM;               // 256*256 f32 (ws needs 512 KB)
  const int B = in_sizes[0] / NQ;

  build_V<<<DIM, 128, 0, stream>>>(w, Vre, Vim);
  const int nblk = (B + MT - 1) / MT;
  qnn_wmma<<<nblk, WAVES * 32, 0, stream>>>(x, Vre, Vim, out, B);
}